// ElasticTransformLayer_84155589198481
// MI455X (gfx1250) — compile-verified
//
#include <hip/hip_runtime.h>
#include <stdint.h>

// Problem shape fixed by reference setup_inputs(): H=2048, W=4096, C=3, 3x3 ctrl grid.
#define HH 2048
#define WW 4096
#define CC 3
#define NC 3
#define TILE 16           // one 16x16 pixel tile per wave (WMMA D-matrix shape)
#define WAVES_PER_BLOCK 8 // 256-thread block = 8 wave32s = 8 tiles in a row

#define ROW_BYTES ((uint32_t)(WW * CC * 4))  // 49152 bytes per image row
#define PIX_BYTES ((uint32_t)(CC * 4))       // 12 bytes per texel

typedef float v2f __attribute__((ext_vector_type(2)));
typedef float v8f __attribute__((ext_vector_type(8)));

struct __attribute__((aligned(4))) F3 { float x, y, z; };

// Keys cubic (a=-0.5) tap weights for fraction t in [0,1): the four taps
// (i0-1, i0, i0+1, i0+2) have closed-form polynomial weights -- branch-free,
// algebraically identical to the piecewise kernel incl. boundaries (t=0 ->
// 0,1,0,0). No compares, no exec-mask changes anywhere (WMMA-safe wave).
__device__ __forceinline__ void keysWeights(float t, float w[4]) {
    float t2 = t * t;
    w[0] = ((-0.5f * t + 1.0f) * t - 0.5f) * t;   // cubw(1+t)
    w[1] = (1.5f * t - 2.5f) * t2 + 1.0f;         // cubw(t)
    w[2] = ((-1.5f * t + 2.0f) * t + 0.5f) * t;   // cubw(1-t)
    w[3] = (0.5f * t - 0.5f) * t2;                // cubw(2-t)
}

// Collapse the 4 clamped cubic taps onto the 3 control points (compare-select
// adds; constant arms -> v_cndmask, no branches).
__device__ __forceinline__ void ctrlWeights(int i, int n_out, float w[3]) {
    float u = (float)(i * (NC - 1)) / (float)(n_out - 1);  // matches jnp op order
    float fi = floorf(u);
    int i0 = (int)fi;
    float tw[4];
    keysWeights(u - fi, tw);
    w[0] = 0.0f; w[1] = 0.0f; w[2] = 0.0f;
#pragma unroll
    for (int k = 0; k < 4; ++k) {
        int idx = i0 - 1 + k;
        idx = idx < 0 ? 0 : (idx > NC - 1 ? NC - 1 : idx);
        w[0] += (idx == 0) ? tw[k] : 0.0f;
        w[1] += (idx == 1) ? tw[k] : 0.0f;
        w[2] += (idx == 2) ? tw[k] : 0.0f;
    }
}

// base + zext(u32 byte offset): selects the SADDR + 32-bit voffset global-load
// form (no per-lane 64-bit address arithmetic; image is 96MB < 2^32).
__device__ __forceinline__ const F3* tapPtr(const float* __restrict__ img, uint32_t byteOff) {
    return (const F3*)((const char*)img + (size_t)byteOff);
}

// Bicubic 4x4 clamped-tap gather at (py+dy, px+dx), NT store of 3 channels.
__device__ __forceinline__ void bicubicSample(
    const float* __restrict__ img, float* __restrict__ out,
    int py, int px, float dy, float dx)
{
    const float yy = (float)py + dy;
    const float xx = (float)px + dx;
    const float fy = floorf(yy), fx = floorf(xx);
    const int iy0 = (int)fy, ix0 = (int)fx;

    float wy[4], wx[4];
    keysWeights(yy - fy, wy);
    keysWeights(xx - fx, wx);

    // Clamped tap offsets as unsigned byte offsets (u24 muls; clamps -> v_med3).
    uint32_t ryo[4], rxo[4];
#pragma unroll
    for (int k = 0; k < 4; ++k) {
        int y = iy0 - 1 + k;
        y = y < 0 ? 0 : (y > HH - 1 ? HH - 1 : y);
        ryo[k] = (uint32_t)y * ROW_BYTES;
        int x = ix0 - 1 + k;
        x = x < 0 ? 0 : (x > WW - 1 ? WW - 1 : x);
        rxo[k] = (uint32_t)x * PIX_BYTES;
    }

    float a0 = 0.0f, a1 = 0.0f, a2 = 0.0f;
#pragma unroll
    for (int ky = 0; ky < 4; ++ky) {
        const uint32_t rowB = ryo[ky];
        float r0 = 0.0f, r1 = 0.0f, r2 = 0.0f;
#pragma unroll
        for (int kx = 0; kx < 4; ++kx) {
            F3 v = *tapPtr(img, rowB + rxo[kx]);  // one b96, saddr+voffset
            r0 = fmaf(wx[kx], v.x, r0);
            r1 = fmaf(wx[kx], v.y, r1);
            r2 = fmaf(wx[kx], v.z, r2);
        }
        a0 = fmaf(wy[ky], r0, a0);
        a1 = fmaf(wy[ky], r1, a1);
        a2 = fmaf(wy[ky], r2, a2);
    }

    // Non-temporal stores: the 96MB output stream must not evict the
    // L2-resident input (input fits the 192MB L2; 16x tap reuse lives there).
    const uint32_t oe = ((uint32_t)py * WW + (uint32_t)px) * CC;  // element idx
    float* op = (float*)((char*)out + (size_t)(oe * 4u));
    __builtin_nontemporal_store(a0, op + 0);
    __builtin_nontemporal_store(a1, op + 1);
    __builtin_nontemporal_store(a2, op + 2);
}

__global__ __launch_bounds__(256) void elastic_warp_kernel(
    const float* __restrict__ img,   // (H, W, C) fp32
    const float* __restrict__ dc,    // (2, 3, 3) fp32 control displacements
    float* __restrict__ out)         // (H, W, C) fp32
{
    const int lane   = threadIdx.x & 31;
    const int waveId = threadIdx.x >> 5;
    const int tileX  = (blockIdx.x * WAVES_PER_BLOCK + waveId) * TILE;
    const int tileY  = blockIdx.y * TILE;
    const int n      = lane & 15;        // row (for A) / column (for B, D) index
    const bool hi    = lane >= 16;       // K-half select per ISA 16x4 f32 layout

    // Wave-uniform reads -> scalar loads into SGPRs; tiny, constant-cache hot.
    float D[18];
#pragma unroll
    for (int j = 0; j < 18; ++j) D[j] = dc[j] * 5.0f;

    // Spline weights once per lane (shared by this lane's 8 pixels).
    float ByW[3], BxW[3];
    ctrlWeights(tileY + n, HH, ByW);
    ctrlWeights(tileX + n, WW, BxW);

    // T_d[k] = sum_x D[d][k][x] * Bx[x]  -> the (3x16) B operand, per column n.
    float Ty[3], Tx[3];
#pragma unroll
    for (int k = 0; k < 3; ++k) {
        Ty[k] = fmaf(D[k * 3 + 0], BxW[0], fmaf(D[k * 3 + 1], BxW[1], D[k * 3 + 2] * BxW[2]));
        Tx[k] = fmaf(D[9 + k * 3 + 0], BxW[0], fmaf(D[9 + k * 3 + 1], BxW[1], D[9 + k * 3 + 2] * BxW[2]));
    }

    // ISA 7.12.2 16x4 f32 A operand (2 VGPRs): v0 = K0|K2, v1 = K1|K3(=0).
    v2f A;
    A.x = hi ? ByW[2] : ByW[0];
    A.y = hi ? 0.0f   : ByW[1];
    // 4x16 f32 B operand, same K-striping with N = lane&15.
    v2f Bdy, Bdx;
    Bdy.x = hi ? Ty[2] : Ty[0];
    Bdy.y = hi ? 0.0f  : Ty[1];
    Bdx.x = hi ? Tx[2] : Tx[0];
    Bdx.y = hi ? 0.0f  : Tx[1];

    // disp tile (16x16) = By_tile (16x4) x (D_d . Bx_tile^T) (4x16) on the
    // matrix unit: two v_wmma_f32_16x16x4_f32 per tile.
    v8f zeroc = {};
    v8f dyv = __builtin_amdgcn_wmma_f32_16x16x4_f32(
        false, A, false, Bdy, (short)0, zeroc, false, false);
    v8f dxv = __builtin_amdgcn_wmma_f32_16x16x4_f32(
        false, A, false, Bdx, (short)0, zeroc, false, false);

    // D-matrix layout: element k of lane L is pixel (M = k + (hi?8:0), N = L&15).
    const int px    = tileX + n;
    const int yBase = tileY + (hi ? 8 : 0);

#pragma unroll
    for (int k = 0; k < 8; ++k) {
        bicubicSample(img, out, yBase + k, px, dyv[k], dxv[k]);
    }
}

extern "C" void kernel_launch(void* const* d_in, const int* in_sizes, int n_in,
                              void* d_out, int out_size, void* d_ws, size_t ws_size,
                              hipStream_t stream) {
    (void)in_sizes; (void)n_in; (void)out_size; (void)d_ws; (void)ws_size;
    const float* img = (const float*)d_in[0];   // (2048, 4096, 3) fp32
    const float* dc  = (const float*)d_in[1];   // (2, 3, 3) fp32
    float* out = (float*)d_out;                 // (2048, 4096, 3) fp32

    dim3 block(256, 1, 1);
    dim3 grid(WW / (TILE * WAVES_PER_BLOCK), HH / TILE, 1);  // (32, 128)
    elastic_warp_kernel<<<grid, block, 0, stream>>>(img, dc, out);
}